// GINEConvEncoder_5231270167281
// MI455X (gfx1250) — compile-verified
//
#include <hip/hip_runtime.h>

typedef float v2f __attribute__((ext_vector_type(2)));
typedef float v8f __attribute__((ext_vector_type(8)));

#define BGR   2048
#define NNODE 64
#define KTOP  52
#define TSUB  16
#define DIM   128
#define D2    256
#define HD    256
#define H2    512
#define EPG   256
#define BN    (BGR * NNODE)
#define ETOT  (BGR * EPG)

__device__ __forceinline__ v8f wmma_f32(v2f a, v2f b, v8f c) {
    return __builtin_amdgcn_wmma_f32_16x16x4_f32(false, a, false, b, (short)0, c,
                                                 false, false);
}

__global__ __launch_bounds__(256) void k0_init(float* p) {
    p[blockIdx.x * 256 + threadIdx.x] = 0.0f;
}

// One-shot transpose fc_w [256,512] -> fc_wT [512,256] so k4's B-fragment
// reads are lane-coalesced.
__global__ __launch_bounds__(256) void k2b_transpose(const float* __restrict__ fc_w,
                                                     float* __restrict__ fc_wT) {
    int idx = blockIdx.x * 256 + threadIdx.x;  // 512 blocks -> 131072 elements
    int o = idx >> 9;
    int k = idx & 511;
    fc_wT[k * HD + o] = fc_w[idx];
}

// Per-graph: embed -> LDS, edge agg in LDS, GEMM1 (WMMA f32), BN stats.
// Wave owns 2 col-tiles, 4 row-tile accumulators: W1 read exactly once/block.
__global__ __launch_bounds__(256) void k1_embed_agg_gemm1(
    const int* __restrict__ tokens, const int* __restrict__ node_type,
    const int* __restrict__ edge_index, const float* __restrict__ st_emb,
    const float* __restrict__ nt_emb, const float* __restrict__ W1,
    const float* __restrict__ b1, float* __restrict__ h1,
    float* __restrict__ g_colsum, float* __restrict__ g_colsq) {
    __shared__ float xs[NNODE * DIM];   // 32 KB: x, then h = x + agg
    __shared__ float agg[NNODE * DIM];  // 32 KB
    __shared__ float cs_s[D2];
    __shared__ float cq_s[D2];

    const int b = blockIdx.x;
    const int tid = threadIdx.x;
    const int wid = tid >> 5;
    const int lane = tid & 31;
    const int half = lane >> 4;
    const int l16 = lane & 15;
    const long gbase = (long)b * NNODE;

    // Phase 1: subtoken-mean embedding + node-type embedding. Wave per 8 nodes,
    // each lane owns a float4 slice of the 128-dim row.
    for (int i = 0; i < 8; ++i) {
        int n = wid * 8 + i;
        long row = gbase + n;
        const int* trow = tokens + row * TSUB;
        float4 acc = make_float4(0.f, 0.f, 0.f, 0.f);
        int npart = 0;
        for (int tt = 0; tt < TSUB; ++tt) {
            int tok = trow[tt];
            npart += (tok != 0);
            const float4 e = *(const float4*)(st_emb + (long)tok * DIM + lane * 4);
            acc.x += e.x; acc.y += e.y; acc.z += e.z; acc.w += e.w;
        }
        if (npart > 0) {
            float inv = 1.0f / (float)npart;
            acc.x *= inv; acc.y *= inv; acc.z *= inv; acc.w *= inv;
        }
        const float4 ne =
            *(const float4*)(nt_emb + (long)node_type[row] * DIM + lane * 4);
        acc.x += ne.x; acc.y += ne.y; acc.z += ne.z; acc.w += ne.w;
        *(float4*)(xs + n * DIM + lane * 4) = acc;
    }
    for (int i = tid; i < NNODE * DIM; i += 256) agg[i] = 0.0f;
    if (tid < D2) { cs_s[tid] = 0.0f; cq_s[tid] = 0.0f; }
    __syncthreads();

    // Phase 2: GINE message aggregation entirely in LDS (ds_add_f32 atomics).
    const int* srcp = edge_index;
    const int* dstp = edge_index + ETOT;
    for (int e = wid; e < EPG; e += 8) {
        int eg = b * EPG + e;
        int s = srcp[eg] - b * NNODE;
        int d = dstp[eg] - b * NNODE;
#pragma unroll
        for (int j = 0; j < 4; ++j) {
            float v = xs[s * DIM + lane * 4 + j];
            v = v > 0.0f ? v : 0.0f;
            atomicAdd(&agg[d * DIM + lane * 4 + j], v);
        }
    }
    __syncthreads();
    for (int i = tid; i < NNODE * DIM; i += 256) xs[i] += agg[i];
    __syncthreads();

    // Phase 3: h1 = h @ W1 + b1 via V_WMMA_F32_16X16X4_F32; fused BN stats.
    // B fragment shared by 4 independent WMMAs (4 row-tile accumulators).
#pragma unroll 1
    for (int q = 0; q < 2; ++q) {
        const int n0 = (wid * 2 + q) * 16;
        v8f acc[4] = {{}, {}, {}, {}};
        for (int k0 = 0; k0 < DIM; k0 += 4) {
            v2f bb;
            const float* bp = W1 + (k0 + 2 * half) * D2 + n0 + l16;
            bb[0] = bp[0];
            bb[1] = bp[D2];
#pragma unroll
            for (int rt = 0; rt < 4; ++rt) {
                v2f a;
                const float* ap = xs + (rt * 16 + l16) * DIM + k0 + 2 * half;
                a[0] = ap[0];
                a[1] = ap[1];
                acc[rt] = wmma_f32(a, bb, acc[rt]);
            }
        }
        float bias = b1[n0 + l16];
        float ls = 0.0f, lq = 0.0f;
#pragma unroll
        for (int rt = 0; rt < 4; ++rt) {
#pragma unroll
            for (int r = 0; r < 8; ++r) {
                float v = acc[rt][r] + bias;
                int row = rt * 16 + r + 8 * half;
                h1[(gbase + row) * D2 + n0 + l16] = v;
                ls += v;
                lq += v * v;
            }
        }
        ls += __shfl_xor(ls, 16, 32);
        lq += __shfl_xor(lq, 16, 32);
        if (half == 0) {
            atomicAdd(&cs_s[n0 + l16], ls);
            atomicAdd(&cq_s[n0 + l16], lq);
        }
    }
    __syncthreads();
    if (tid < D2) {
        atomicAdd(&g_colsum[tid], cs_s[tid]);
        atomicAdd(&g_colsq[tid], cq_s[tid]);
    }
}

// BN stats + score pre-folding: w2t = W2 @ topk_w, b2t = b2 . topk_w, 1/||topk_w||.
__global__ __launch_bounds__(256) void k2_stats(
    const float* __restrict__ g_colsum, const float* __restrict__ g_colsq,
    const float* __restrict__ W2, const float* __restrict__ b2,
    const float* __restrict__ topk_w, float* __restrict__ mu,
    float* __restrict__ rstd, float* __restrict__ w2t, float* __restrict__ misc) {
    int t = threadIdx.x;
    float m = g_colsum[t] * (1.0f / (float)BN);
    float v = g_colsq[t] * (1.0f / (float)BN) - m * m;
    mu[t] = m;
    rstd[t] = rsqrtf(v + 1e-5f);
    float s = 0.0f;
    for (int c = 0; c < HD; ++c) s += W2[t * HD + c] * topk_w[c];
    w2t[t] = s;
    if (t == 0) {
        float nn = 0.0f, bt = 0.0f;
        for (int c = 0; c < HD; ++c) {
            nn += topk_w[c] * topk_w[c];
            bt += b2[c] * topk_w[c];
        }
        misc[0] = rsqrtf(nn);
        misc[1] = bt;
    }
}

// Per-graph: normalize+ReLU, scores, top-K select, GEMM2 (WMMA, B shared across
// 4 row-tile accumulators) with fused gated max/mean pooling -> cat to ws.
__global__ __launch_bounds__(256) void k3_norm_gemm2_pool(
    const float* __restrict__ h1, const float* __restrict__ gamma,
    const float* __restrict__ beta, const float* __restrict__ W2,
    const float* __restrict__ b2, const float* __restrict__ mu,
    const float* __restrict__ rstd, const float* __restrict__ w2t,
    const float* __restrict__ misc, float* __restrict__ catp) {
    __shared__ float hn[NNODE * D2];  // 64 KB
    __shared__ float sc_s[NNODE];
    __shared__ float gate_s[NNODE];
    __shared__ int sel_s[NNODE];

    const int b = blockIdx.x;
    const int tid = threadIdx.x;
    const int wid = tid >> 5;
    const int lane = tid & 31;
    const int half = lane >> 4;
    const int l16 = lane & 15;
    const long gbase = (long)b * NNODE;

    // Phase 1: hn = relu(BN(h1))
    for (int i = tid; i < NNODE * D2; i += 256) {
        int c = i & (D2 - 1);
        float v = h1[gbase * D2 + i];
        v = (v - mu[c]) * rstd[c] * gamma[c] + beta[c];
        hn[i] = v > 0.0f ? v : 0.0f;
    }
    __syncthreads();

    // Phase 2: projection scores (pre-folded through GEMM2).
    if (tid < NNODE) {
        float s = 0.0f;
        for (int k = 0; k < D2; ++k) s += hn[tid * D2 + k] * w2t[k];
        sc_s[tid] = (s + misc[1]) * misc[0];
    }
    __syncthreads();
    // Phase 3: rank-based top-K (matches top_k value-then-index ordering).
    if (tid < NNODE) {
        float my = sc_s[tid];
        int rank = 0;
        for (int j = 0; j < NNODE; ++j) {
            float o = sc_s[j];
            rank += (o > my) || (o == my && j < tid);
        }
        int sel = rank < KTOP;
        sel_s[tid] = sel;
        gate_s[tid] = sel ? tanhf(my) : 0.0f;
    }
    __syncthreads();

    // Phase 4: GEMM2 via WMMA; wave owns col-tiles 2w,2w+1, 4 row-tile
    // accumulators share each B fragment; pooling fused from the fragments.
#pragma unroll 1
    for (int q = 0; q < 2; ++q) {
        const int n0 = (wid * 2 + q) * 16;
        v8f acc[4] = {{}, {}, {}, {}};
        for (int k0 = 0; k0 < D2; k0 += 4) {
            v2f bb;
            const float* bp = W2 + (k0 + 2 * half) * HD + n0 + l16;
            bb[0] = bp[0];
            bb[1] = bp[HD];
#pragma unroll
            for (int rt = 0; rt < 4; ++rt) {
                v2f a;
                const float* ap = hn + (rt * 16 + l16) * D2 + k0 + 2 * half;
                a[0] = ap[0];
                a[1] = ap[1];
                acc[rt] = wmma_f32(a, bb, acc[rt]);
            }
        }
        float bias = b2[n0 + l16];
        float cmax = -3.402823e38f;
        float csum = 0.0f;
#pragma unroll
        for (int rt = 0; rt < 4; ++rt) {
#pragma unroll
            for (int r = 0; r < 8; ++r) {
                int row = rt * 16 + r + 8 * half;
                float gv = gate_s[row] * (acc[rt][r] + bias);
                int se = sel_s[row];
                csum += se ? gv : 0.0f;
                cmax = se ? fmaxf(cmax, gv) : cmax;
            }
        }
        csum += __shfl_xor(csum, 16, 32);
        cmax = fmaxf(cmax, __shfl_xor(cmax, 16, 32));
        if (half == 0) {
            catp[(long)b * H2 + n0 + l16] = cmax;
            catp[(long)b * H2 + HD + n0 + l16] = csum * (1.0f / (float)KTOP);
        }
    }
}

// Batched final linear: out[2048,256] = cat[2048,512] @ fc_wT + fc_b, via WMMA.
__global__ __launch_bounds__(256) void k4_fc(const float* __restrict__ catp,
                                             const float* __restrict__ fc_wT,
                                             const float* __restrict__ fc_b,
                                             float* __restrict__ out) {
    __shared__ float as[16 * H2];  // 32 KB row-tile of cat

    const int tid = threadIdx.x;
    const int wid = tid >> 5;
    const int lane = tid & 31;
    const int half = lane >> 4;
    const int l16 = lane & 15;
    const int g0 = blockIdx.x * 16;  // 128 blocks, 16 graphs each

    for (int i = tid; i < 16 * H2; i += 256) as[i] = catp[(long)g0 * H2 + i];
    __syncthreads();

#pragma unroll 1
    for (int q = 0; q < 2; ++q) {
        const int n0 = (wid * 2 + q) * 16;
        v8f acc = {};
        for (int k0 = 0; k0 < H2; k0 += 4) {
            v2f a, bb;
            const float* ap = as + l16 * H2 + k0 + 2 * half;
            a[0] = ap[0];
            a[1] = ap[1];
            const float* bp = fc_wT + (k0 + 2 * half) * HD + n0 + l16;
            bb[0] = bp[0];
            bb[1] = bp[HD];
            acc = wmma_f32(a, bb, acc);
        }
        float bias = fc_b[n0 + l16];
#pragma unroll
        for (int r = 0; r < 8; ++r) {
            out[(long)(g0 + r + 8 * half) * HD + n0 + l16] = acc[r] + bias;
        }
    }
}

extern "C" void kernel_launch(void* const* d_in, const int* in_sizes, int n_in,
                              void* d_out, int out_size, void* d_ws, size_t ws_size,
                              hipStream_t stream) {
    (void)in_sizes; (void)n_in; (void)out_size; (void)ws_size;
    const int* tokens = (const int*)d_in[0];
    const int* node_type = (const int*)d_in[1];
    const int* edge_index = (const int*)d_in[2];
    const float* st_emb = (const float*)d_in[3];
    const float* nt_emb = (const float*)d_in[4];
    const float* W1 = (const float*)d_in[5];
    const float* b1 = (const float*)d_in[6];
    const float* gamma = (const float*)d_in[7];
    const float* beta = (const float*)d_in[8];
    const float* W2 = (const float*)d_in[9];
    const float* b2 = (const float*)d_in[10];
    const float* topk_w = (const float*)d_in[11];
    const float* fc_w = (const float*)d_in[12];
    const float* fc_b = (const float*)d_in[13];

    float* ws = (float*)d_ws;
    float* h1 = ws;                                 // [BN, 256] = 128 MB
    float* colsum = ws + (size_t)BN * D2;           // [256]
    float* colsq = colsum + D2;                     // [256]
    float* mu = colsq + D2;                         // [256]
    float* rstd = mu + D2;                          // [256]
    float* w2t = rstd + D2;                         // [256]
    float* misc = w2t + D2;                         // [8]
    float* fc_wT = misc + 8;                        // [512, 256]
    float* catp = fc_wT + (size_t)H2 * HD;          // [2048, 512]
    float* out = (float*)d_out;

    k0_init<<<2, 256, 0, stream>>>(colsum);  // zeros colsum+colsq (contiguous 512)
    k2b_transpose<<<512, 256, 0, stream>>>(fc_w, fc_wT);
    k1_embed_agg_gemm1<<<BGR, 256, 0, stream>>>(tokens, node_type, edge_index,
                                                st_emb, nt_emb, W1, b1, h1,
                                                colsum, colsq);
    k2_stats<<<1, 256, 0, stream>>>(colsum, colsq, W2, b2, topk_w, mu, rstd, w2t,
                                    misc);
    k3_norm_gemm2_pool<<<BGR, 256, 0, stream>>>(h1, gamma, beta, W2, b2, mu, rstd,
                                                w2t, misc, catp);
    k4_fc<<<128, 256, 0, stream>>>(catp, fc_wT, fc_b, out);
}